// CFGAN_9526237462749
// MI455X (gfx1250) — compile-verified
//
#include <hip/hip_runtime.h>
#include <hip/hip_bf16.h>
#include <math.h>

// CFGAN generator stack for MI455X (gfx1250, wave32).
// Strategy: per net, 5 fused passes with BN batch-stat reductions at pass
// boundaries (kernel launches on one stream = grid sync). Middle Linear(8,8)
// layers use V_WMMA_F32_16X16X4_F32 (two K=4 WMMAs per 16-row tile).
// h ping-pong buffers (16MB) stay L2-resident (192MB L2), HBM sees mostly
// the 230MB input + 23MB output -> bandwidth-optimal for this DAG.

#define BATCH    524288
#define BN_EPS   0.8f
#define SLOPE    0.2f

typedef float v2f __attribute__((ext_vector_type(2)));
typedef float v8f __attribute__((ext_vector_type(8)));

struct FeatSpec { short code[20]; };   // code<1000: Z offset (slice*10+j); >=1000: out column (code-1000)

__device__ __forceinline__ float lrelu(float x) { return x >= 0.f ? x : SLOPE * x; }

// ---------------- Pass 1: gather input, Linear(in,8)+lrelu, Linear(8,8), stats of h1 ----------------
template <int NIN>
__global__ __launch_bounds__(256) void cfgan_pass1(
    const float* __restrict__ Zin, const float* __restrict__ outBuf, FeatSpec fs,
    const float* __restrict__ W0, const float* __restrict__ b0,
    const float* __restrict__ W1, const float* __restrict__ b1,
    float* __restrict__ hout, float* __restrict__ stats)
{
    __shared__ float sW0[NIN * 8], sb0[8], sW1[64], sb1[8], sacc[16];
    const int tid = threadIdx.x;
    if (tid < NIN * 8) sW0[tid] = W0[tid];
    if (tid < 64)      sW1[tid] = W1[tid];
    if (tid < 8)  { sb0[tid] = b0[tid]; sb1[tid] = b1[tid]; }
    if (tid < 16) sacc[tid] = 0.f;
    __syncthreads();

    float s[8], ss[8];
#pragma unroll
    for (int n = 0; n < 8; n++) { s[n] = 0.f; ss[n] = 0.f; }

    const int stride = gridDim.x * blockDim.x;
    for (int row = blockIdx.x * blockDim.x + tid; row < BATCH; row += stride) {
        float xin[NIN];
#pragma unroll
        for (int k = 0; k < NIN; k++) {
            const int c = fs.code[k];
            xin[k] = (c < 1000) ? Zin[(size_t)row * 110 + c]
                                : outBuf[(size_t)row * 11 + (c - 1000)];
        }
        float x0[8];
#pragma unroll
        for (int n = 0; n < 8; n++) {
            float a = sb0[n];
#pragma unroll
            for (int k = 0; k < NIN; k++) a = fmaf(xin[k], sW0[k * 8 + n], a);
            x0[n] = lrelu(a);
        }
#pragma unroll
        for (int n = 0; n < 8; n++) {
            float a = sb1[n];
#pragma unroll
            for (int k = 0; k < 8; k++) a = fmaf(x0[k], sW1[k * 8 + n], a);
            hout[(size_t)row * 8 + n] = a;
            s[n] += a; ss[n] += a * a;
        }
    }
#pragma unroll
    for (int n = 0; n < 8; n++) { atomicAdd(&sacc[n], s[n]); atomicAdd(&sacc[8 + n], ss[n]); }
    __syncthreads();
    if (tid < 16) atomicAdd(&stats[tid], sacc[tid]);
}

// ------------- Middle passes: BN(statsIn)+lrelu, Linear(8,8) via WMMA f32 16x16x4, stats of h_next -------------
// A layout (ISA 7.12.2, 32-bit A 16x4): lane m=lane&15 is row M; half=lane>>4 selects K pair
//   {2*half, 2*half+1}. Two K=4 tiles cover K=8. B 4x16: VGPR j holds row K=j+2*half, N=lane&15.
// D 16x16: VGPR v holds row v+8*half, column N=lane&15 -> per-lane column partial sums for BN stats.
__global__ __launch_bounds__(256) void cfgan_passmid(
    const float* __restrict__ hin, float* __restrict__ hout,
    const float* __restrict__ statsIn, float* __restrict__ statsOut,
    const float* __restrict__ W, const float* __restrict__ bias,
    const float* __restrict__ gamma, const float* __restrict__ beta,
    float invN)
{
    __shared__ float sacc[16];
    const int tid = threadIdx.x;
    if (tid < 16) sacc[tid] = 0.f;
    __syncthreads();

    const int lane = tid & 31;
    const int m    = lane & 15;          // row within tile (A), column (B/D)
    const int half = lane >> 4;
    const int wave = blockIdx.x * (blockDim.x >> 5) + (tid >> 5);
    const int r0   = wave << 4;          // 16-row tile base
    const int k0   = 2 * half;

    const float* hrow = hin + (size_t)(r0 + m) * 8;
    float y[4];
#pragma unroll
    for (int i = 0; i < 4; i++) {
        const int k    = (i < 2) ? (k0 + i) : (4 + k0 + (i - 2));
        const float mn = statsIn[k] * invN;
        const float vr = statsIn[8 + k] * invN - mn * mn;      // biased variance
        const float rs = rsqrtf(vr + BN_EPS);
        const float sc = gamma[k] * rs;
        const float sh = beta[k] - mn * sc;
        y[i] = lrelu(fmaf(hrow[k], sc, sh));
    }
    v2f a0 = { y[0], y[1] };   // K = k0, k0+1
    v2f a1 = { y[2], y[3] };   // K = 4+k0, 4+k0+1

    const bool cw = (m < 8);
    v2f bB0 = { cw ? W[(k0 + 0) * 8 + m] : 0.f,  cw ? W[(k0 + 1) * 8 + m] : 0.f };
    v2f bB1 = { cw ? W[(4 + k0 + 0) * 8 + m] : 0.f, cw ? W[(4 + k0 + 1) * 8 + m] : 0.f };

    const float cb = cw ? bias[m] : 0.f;
    v8f acc = { cb, cb, cb, cb, cb, cb, cb, cb };
    acc = __builtin_amdgcn_wmma_f32_16x16x4_f32(false, a0, false, bB0, (short)0, acc, false, false);
    acc = __builtin_amdgcn_wmma_f32_16x16x4_f32(false, a1, false, bB1, (short)0, acc, false, false);

    float s = 0.f, ssq = 0.f;
#pragma unroll
    for (int v = 0; v < 8; v++) { const float d = acc[v]; s += d; ssq += d * d; }

    if (cw) {
#pragma unroll
        for (int v = 0; v < 8; v++)
            hout[(size_t)(r0 + v + 8 * half) * 8 + m] = acc[v];
        atomicAdd(&sacc[m], s);
        atomicAdd(&sacc[8 + m], ssq);
    }
    __syncthreads();
    if (tid < 16) atomicAdd(&statsOut[tid], sacc[tid]);
}

// ---------------- Pass 5: BN+lrelu, Linear(8,1), tanh, scatter to output column ----------------
__global__ __launch_bounds__(256) void cfgan_pass5(
    const float* __restrict__ hin, const float* __restrict__ statsIn,
    const float* __restrict__ W5, const float* __restrict__ b5,
    const float* __restrict__ gamma, const float* __restrict__ beta,
    float* __restrict__ outBuf, int outCol, float invN)
{
    float sc[8], sh[8], w[8];
#pragma unroll
    for (int k = 0; k < 8; k++) {
        const float mn = statsIn[k] * invN;
        const float vr = statsIn[8 + k] * invN - mn * mn;
        const float rs = rsqrtf(vr + BN_EPS);
        sc[k] = gamma[k] * rs;
        sh[k] = beta[k] - mn * sc[k];
        w[k]  = W5[k];
    }
    const float bb = b5[0];
    const int stride = gridDim.x * blockDim.x;
    for (int row = blockIdx.x * blockDim.x + threadIdx.x; row < BATCH; row += stride) {
        const float* hr = hin + (size_t)row * 8;
        float a = bb;
#pragma unroll
        for (int k = 0; k < 8; k++)
            a = fmaf(lrelu(fmaf(hr[k], sc[k], sh[k])), w[k], a);
        outBuf[(size_t)row * 11 + outCol] = tanhf(a);
    }
}

__global__ void cfgan_zero(float* p, int n) {
    const int i = blockIdx.x * blockDim.x + threadIdx.x;
    if (i < n) p[i] = 0.f;
}

// ---------------------------------------------------------------------------------------------
extern "C" void kernel_launch(void* const* d_in, const int* in_sizes, int n_in,
                              void* d_out, int out_size, void* d_ws, size_t ws_size,
                              hipStream_t stream)
{
    const float* Zin = (const float*)d_in[0];
    float* out   = (float*)d_out;
    float* hA    = (float*)d_ws;
    float* hB    = hA + (size_t)BATCH * 8;
    float* stats = hB + (size_t)BATCH * 8;           // 11 nets * 4 layers * 16 floats
    if (ws_size < ((size_t)BATCH * 16 + 704) * sizeof(float)) return;
    const float invN = 1.0f / (float)BATCH;

    // params flattening order: detect insertion-order vs jax-pytree (sorted keys)
    // via second net's W0 size: workclass(14*8=112) vs edu_level(15*8=120).
    const bool srt = (n_in > 21) && (in_sizes[21] == 120);
    static const int POS_SORT[11] = {0, 10, 1, 4, 6, 8, 7, 9, 2, 5, 3};
    // param ids (insertion order): 0 age,1 workclass,2 edu,3 marital,4 occ,
    //                              5 relationship,6 race,7 sex,8 hours,9 native,10 income
    static const int taskParam[11]  = {6, 0, 7, 9, 3, 2, 4, 8, 1, 5, 10}; // note race/sex params swapped per reference
    static const int taskOutCol[11] = {7, 0, 6, 9, 3, 2, 4, 8, 1, 5, 10};
    static const int taskNin[11]    = {10,10,10,10,14,15,15,16,14,15,20};
    static const int taskZslice[11] = {2, 1, 0, 3, 4, 5, 6, 7, 8, 9, 10};
    static const short ex4[11][10] = {
        {0},{0},{0},{0},
        {1006,1000,1007,1009},                                    // ms
        {1006,1000,1007,1009,1003},                               // el
        {1006,1000,1007,1003,1002},                               // oc
        {1006,1000,1007,1009,1003,1002},                          // hw
        {1000,1003,1002,1009},                                    // wc
        {1000,1007,1009,1003,1002},                               // rel
        {1006,1000,1007,1009,1003,1002,1004,1008,1001,1005},      // inc
    };

    cfgan_zero<<<4, 256, 0, stream>>>(stats, 704);

    for (int t = 0; t < 11; t++) {
        const int pid  = taskParam[t];
        const int base = 1 + (srt ? POS_SORT[pid] : pid) * 20;
        const float *W[6], *Bv[6], *G[4], *Bt[4];
        for (int l = 0; l < 6; l++) { W[l]  = (const float*)d_in[base + l];
                                      Bv[l] = (const float*)d_in[base + 6 + l]; }
        const int go = srt ? 16 : 12, bo = srt ? 12 : 16;   // sorted: beta before gamma
        for (int l = 0; l < 4; l++) { G[l]  = (const float*)d_in[base + go + l];
                                      Bt[l] = (const float*)d_in[base + bo + l]; }
        float* sb = stats + t * 64;

        FeatSpec fs{};
        for (int k = 0; k < 10; k++) fs.code[k] = (short)(taskZslice[t] * 10 + k);
        for (int k = 10; k < taskNin[t]; k++) fs.code[k] = ex4[t][k - 10];

        switch (taskNin[t]) {
            case 10: cfgan_pass1<10><<<512,256,0,stream>>>(Zin,out,fs,W[0],Bv[0],W[1],Bv[1],hA,sb); break;
            case 14: cfgan_pass1<14><<<512,256,0,stream>>>(Zin,out,fs,W[0],Bv[0],W[1],Bv[1],hA,sb); break;
            case 15: cfgan_pass1<15><<<512,256,0,stream>>>(Zin,out,fs,W[0],Bv[0],W[1],Bv[1],hA,sb); break;
            case 16: cfgan_pass1<16><<<512,256,0,stream>>>(Zin,out,fs,W[0],Bv[0],W[1],Bv[1],hA,sb); break;
            default: cfgan_pass1<20><<<512,256,0,stream>>>(Zin,out,fs,W[0],Bv[0],W[1],Bv[1],hA,sb); break;
        }
        // 32768 tiles of 16 rows, 8 waves/block -> 4096 blocks
        cfgan_passmid<<<4096,256,0,stream>>>(hA,hB,sb,      sb+16, W[2],Bv[2],G[0],Bt[0],invN);
        cfgan_passmid<<<4096,256,0,stream>>>(hB,hA,sb+16,   sb+32, W[3],Bv[3],G[1],Bt[1],invN);
        cfgan_passmid<<<4096,256,0,stream>>>(hA,hB,sb+32,   sb+48, W[4],Bv[4],G[2],Bt[2],invN);
        cfgan_pass5  <<<512,256,0,stream>>>(hB,sb+48,W[5],Bv[5],G[3],Bt[3],out,taskOutCol[t],invN);
    }
}